// TemperGraph_4389456576808
// MI455X (gfx1250) — compile-verified
//
#include <hip/hip_runtime.h>
#include <math.h>

// ---------------------------------------------------------------------------
// TemperGraph on MI455X (gfx1250, wave32, WMMA).
// Gather-based MoE (bin active rows by temper each hop): ~8 GFLOP instead of
// the reference's dense ~77 GFLOP. All GEMMs (including the 16x13 router
// logits) run on V_WMMA_F32_16X16X4_F32 so routing argmax stays f32-faithful.
// Weights/activations are L2-resident (<<192MB). A-tiles staged into LDS via
// GLOBAL_LOAD_ASYNC_TO_LDS_B128 (ASYNCcnt) with stride 260 (==4 mod 64 banks,
// 16B aligned rows -> conflict-free WMMA fragment reads, b128-capable).
// ---------------------------------------------------------------------------

#define BATCH    2048
#define IN_DIM   1024
#define HID      256
#define ODIM     512
#define NT_T     12      // number of tempers
#define KOPS     3       // ops per temper
#define NHOPS    4
#define TP1      13      // T + 1
#define MAXTILES 140     // <= BATCH/16 + NT_T
#define LSTR     260     // LDS row stride: %64==4 (conflict-free), 16B aligned
#define SXSTR    516     // proj_in LDS stride (512 cols): %64==4, 16B aligned

#define USE_ASYNC_STAGING 1

typedef float v2f __attribute__((ext_vector_type(2)));
typedef float v8f __attribute__((ext_vector_type(8)));

__device__ __forceinline__ v8f wmma4(v2f a, v2f b, v8f c) {
  // D = A(16x4,f32) * B(4x16,f32) + C(16x16,f32)
  return __builtin_amdgcn_wmma_f32_16x16x4_f32(false, a, false, b,
                                               (short)0, c, false, false);
}

// Async global -> LDS 16B copy, tracked by ASYNCcnt (CDNA5 path).
// LDS byte offset comes from generic-pointer truncation (ISA: LDS generic
// addresses carry the wave-relative LDS offset in addr[31:0]).
__device__ __forceinline__ void async_g2l_b128(unsigned lds_off,
                                               const float* gaddr) {
  asm volatile("global_load_async_to_lds_b128 %0, %1, off"
               :: "v"(lds_off), "v"(gaddr) : "memory");
}
__device__ __forceinline__ void wait_async0() {
#if __has_builtin(__builtin_amdgcn_s_wait_asynccnt)
  __builtin_amdgcn_s_wait_asynccnt(0);
#else
  asm volatile("s_wait_asynccnt 0x0" ::: "memory");
#endif
}

// Accumulate a 16x16 output fragment for columns [nt*16, nt*16+16):
//   A (16 x Kdim) row-major in LDS (row stride lda),
//   B (Kdim x ldb) row-major in global (L2-resident weights).
// Fragment layouts per CDNA5 ISA:
//   A:   lane = m + 16*(k>=2), reg r -> k = 4*ks + 2*(lane>>4) + r
//   B:   reg r, lane l -> row k = 4*ks + 2*(l>>4) + r, col n = l&15
//   C/D: reg r, lane l -> row m = r + 8*(l>>4),        col n = l&15
__device__ __forceinline__ v8f gemm16(const float* __restrict__ A, int lda,
                                      const float* __restrict__ Bg, int ldb,
                                      int Kdim, int nt, int lane, v8f c) {
  const int m16 = lane & 15;
  const int hi  = lane >> 4;
  const int n   = nt * 16 + m16;
  #pragma unroll 4
  for (int ks = 0; ks < (Kdim >> 2); ++ks) {
    const int kk = (ks << 2) + (hi << 1);
    v2f a, b;
    a.x = A[m16 * lda + kk];
    a.y = A[m16 * lda + kk + 1];
    b.x = Bg[(size_t)kk * ldb + n];
    b.y = Bg[(size_t)(kk + 1) * ldb + n];
    c = wmma4(a, b, c);
  }
  return c;
}

// ---------------------------------------------------------------------------
// 0) init per-row routing state
__global__ void init_state_kernel(const int* __restrict__ init_t,
                                  int* __restrict__ tempers,
                                  int* __restrict__ done) {
  int n = blockIdx.x * blockDim.x + threadIdx.x;
  if (n < BATCH) { tempers[n] = init_t[n]; done[n] = 0; }
}

// ---------------------------------------------------------------------------
// 1) states = x @ W_in + b_in      [2048x1024]*[1024x256]
//    128 threads = 4 waves; each wave owns 4 column sub-tiles (16 nt total).
//    K=1024 processed in two 512-column LDS chunks.
__global__ __launch_bounds__(128) void proj_in_kernel(
    const float* __restrict__ x, const float* __restrict__ W_in,
    const float* __restrict__ b_in, float* __restrict__ states) {
  __shared__ float sX[16 * SXSTR];
  const int tile = blockIdx.x;
  const int tid  = threadIdx.x;
  const int lane = tid & 31, wave = tid >> 5;
  const int m16  = lane & 15, hi = lane >> 4;

  v8f zero8 = {};
  v8f c[4];
  #pragma unroll
  for (int i = 0; i < 4; ++i) c[i] = zero8;

  for (int chunk = 0; chunk < 2; ++chunk) {
#if USE_ASYNC_STAGING
    for (int idx = tid; idx < 16 * 128; idx += 128) {  // 128 float4 per row
      int m = idx >> 7, c4 = idx & 127;
      async_g2l_b128((unsigned)(uintptr_t)&sX[m * SXSTR + c4 * 4],
                     x + (size_t)(tile * 16 + m) * IN_DIM + chunk * 512 + c4 * 4);
    }
    wait_async0();
#else
    for (int idx = tid; idx < 16 * 128; idx += 128) {
      int m = idx >> 7, c4 = idx & 127;
      *(float4*)&sX[m * SXSTR + c4 * 4] = *(const float4*)(
          x + (size_t)(tile * 16 + m) * IN_DIM + chunk * 512 + c4 * 4);
    }
#endif
    __syncthreads();
    const float* Bg = W_in + (size_t)chunk * 512 * HID;
    #pragma unroll
    for (int i = 0; i < 4; ++i) {
      int nt = wave * 4 + i;
      c[i] = gemm16(sX, SXSTR, Bg, HID, 512, nt, lane, c[i]);
    }
    __syncthreads();
  }
  #pragma unroll
  for (int i = 0; i < 4; ++i) {
    int nt = wave * 4 + i, n = nt * 16 + m16;
    float bias = b_in[n];
    #pragma unroll
    for (int r = 0; r < 8; ++r)
      states[(size_t)(tile * 16 + r + 8 * hi) * HID + n] = c[i][r] + bias;
  }
}

// ---------------------------------------------------------------------------
// 2) Binning: group active rows by temper so WMMA tiles share one expert.
__global__ void bin_zero_kernel(int* counts, int* cursor, int* numTiles) {
  int t = threadIdx.x;
  if (t < NT_T) { counts[t] = 0; cursor[t] = 0; }
  if (t == 0) numTiles[0] = 0;
}

__global__ void bin_count_kernel(const int* __restrict__ tempers,
                                 const int* __restrict__ done,
                                 int* __restrict__ counts) {
  int n = blockIdx.x * blockDim.x + threadIdx.x;
  if (n < BATCH && !done[n]) atomicAdd(&counts[tempers[n]], 1);
}

__global__ void bin_tiles_kernel(const int* __restrict__ counts,
                                 int* __restrict__ cursor,
                                 int* __restrict__ tileT,
                                 int* __restrict__ tileStart,
                                 int* __restrict__ tileLimit,
                                 int* __restrict__ numTiles) {
  if (blockIdx.x == 0 && threadIdx.x == 0) {
    int off = 0, ntile = 0;
    for (int t = 0; t < NT_T; ++t) {
      int cnt = counts[t];
      cursor[t] = off;
      int end = off + cnt;
      for (int s = off; s < end; s += 16) {
        tileT[ntile] = t; tileStart[ntile] = s; tileLimit[ntile] = end;
        ++ntile;
      }
      off = end;
    }
    numTiles[0] = ntile;
  }
}

__global__ void bin_scatter_kernel(const int* __restrict__ tempers,
                                   const int* __restrict__ done,
                                   int* __restrict__ cursor,
                                   int* __restrict__ rowIdx) {
  int n = blockIdx.x * blockDim.x + threadIdx.x;
  if (n < BATCH && !done[n]) {
    int p = atomicAdd(&cursor[tempers[n]], 1);
    rowIdx[p] = n;
  }
}

// ---------------------------------------------------------------------------
// 3) Expert MLP + router for one hop. One block = one 16-row tile, one temper.
__global__ __launch_bounds__(128) void expert_kernel(
    const float* __restrict__ W1, const float* __restrict__ b1,
    const float* __restrict__ W2, const float* __restrict__ b2,
    const float* __restrict__ Wr, const float* __restrict__ br,
    const float* __restrict__ op_noise, const float* __restrict__ route_u,
    float* __restrict__ states, int* __restrict__ tempers,
    int* __restrict__ done,
    const int* __restrict__ rowIdx, const int* __restrict__ tileT,
    const int* __restrict__ tileStart, const int* __restrict__ tileLimit,
    const int* __restrict__ numTiles, int hop) {
  __shared__ float sS[16 * LSTR];     // input states tile (A of layer 1)
  __shared__ float sH[16 * LSTR];     // h1 for current k (A of layer 2)
  __shared__ float sOut[16 * LSTR];   // weighted expert output
  __shared__ float sLog[16 * TP1];
  __shared__ float sW[KOPS];
  __shared__ int   sRow[16];

  const int b = blockIdx.x;
  if (b >= numTiles[0]) return;       // uniform per block
  const int t     = tileT[b];
  const int start = tileStart[b];
  const int limit = tileLimit[b];
  const int tid   = threadIdx.x;
  const int lane  = tid & 31, wave = tid >> 5;
  const int m16   = lane & 15, hi = lane >> 4;

  if (tid < 16) {
    int s = start + tid;
    sRow[tid] = (s < limit) ? rowIdx[s] : -1;
  }
  if (tid == 0) {  // softmax over op_noise[hop, t, :]
    float a0 = op_noise[(hop * NT_T + t) * KOPS + 0];
    float a1 = op_noise[(hop * NT_T + t) * KOPS + 1];
    float a2 = op_noise[(hop * NT_T + t) * KOPS + 2];
    float mx = fmaxf(a0, fmaxf(a1, a2));
    float e0 = expf(a0 - mx), e1 = expf(a1 - mx), e2 = expf(a2 - mx);
    float inv = 1.0f / (e0 + e1 + e2);
    sW[0] = e0 * inv; sW[1] = e1 * inv; sW[2] = e2 * inv;
  }
  __syncthreads();

  // stage states tile (async b128 where possible; zeros for pad rows)
  for (int idx = tid; idx < 16 * (HID / 4); idx += 128) {
    int m = idx >> 6, c4 = idx & 63;
    int g = sRow[m];
    if (g >= 0) {
#if USE_ASYNC_STAGING
      async_g2l_b128((unsigned)(uintptr_t)&sS[m * LSTR + c4 * 4],
                     states + (size_t)g * HID + c4 * 4);
#else
      *(float4*)&sS[m * LSTR + c4 * 4] =
          *(const float4*)(states + (size_t)g * HID + c4 * 4);
#endif
    } else {
      *(float4*)&sS[m * LSTR + c4 * 4] = make_float4(0.f, 0.f, 0.f, 0.f);
    }
  }
#if USE_ASYNC_STAGING
  wait_async0();
#endif
  __syncthreads();

  float acc[4][8];
  #pragma unroll
  for (int i = 0; i < 4; ++i)
    #pragma unroll
    for (int r = 0; r < 8; ++r) acc[i][r] = 0.0f;

  for (int k = 0; k < KOPS; ++k) {
    const float* W1k = W1 + (size_t)(t * KOPS + k) * HID * HID;
    const float* b1k = b1 + (t * KOPS + k) * HID;
    const float* W2k = W2 + (size_t)(t * KOPS + k) * HID * HID;
    const float* b2k = b2 + (t * KOPS + k) * HID;
    // layer 1: sH = relu(sS @ W1[t,k] + b1[t,k])
    #pragma unroll
    for (int i = 0; i < 4; ++i) {
      int nt = wave * 4 + i, n = nt * 16 + m16;
      // warm WGP$ with the W2 panel needed right after (1 line per row slice)
      #pragma unroll
      for (int p = 0; p < 8; ++p)
        __builtin_prefetch(W2k + (size_t)(p * 32 + lane) * HID + nt * 16, 0, 3);
      v8f c = {};
      c = gemm16(sS, LSTR, W1k, HID, HID, nt, lane, c);
      float bias = b1k[n];
      #pragma unroll
      for (int r = 0; r < 8; ++r) {
        float v = c[r] + bias;
        sH[(r + 8 * hi) * LSTR + n] = v > 0.0f ? v : 0.0f;
      }
    }
    __syncthreads();
    // layer 2: acc += w[k] * relu(sH @ W2[t,k] + b2[t,k])
    float wk = sW[k];
    #pragma unroll
    for (int i = 0; i < 4; ++i) {
      int nt = wave * 4 + i, n = nt * 16 + m16;
      if (k + 1 < KOPS) {  // warm next expert's W1 panel
        const float* W1n = W1 + (size_t)(t * KOPS + k + 1) * HID * HID;
        #pragma unroll
        for (int p = 0; p < 8; ++p)
          __builtin_prefetch(W1n + (size_t)(p * 32 + lane) * HID + nt * 16, 0, 3);
      }
      v8f c = {};
      c = gemm16(sH, LSTR, W2k, HID, HID, nt, lane, c);
      float bias = b2k[n];
      #pragma unroll
      for (int r = 0; r < 8; ++r) {
        float v = c[r] + bias;
        acc[i][r] += wk * (v > 0.0f ? v : 0.0f);
      }
    }
    __syncthreads();  // sH reused next k
  }

  #pragma unroll
  for (int i = 0; i < 4; ++i) {
    int nt = wave * 4 + i, n = nt * 16 + m16;
    #pragma unroll
    for (int r = 0; r < 8; ++r)
      sOut[(r + 8 * hi) * LSTR + n] = acc[i][r];
  }
  __syncthreads();

  // router logits: one WMMA column sub-tile (16x256 @ 256x13), wave 0 only.
  // Columns clamped to TP1-1 so padded lanes never read past Wr's tail.
  if (wave == 0) {
    const float* WrT = Wr + (size_t)t * HID * TP1;
    const int n = (m16 < TP1) ? m16 : (TP1 - 1);
    v8f c = {};
    #pragma unroll 4
    for (int ks = 0; ks < (HID >> 2); ++ks) {
      const int kk = (ks << 2) + (hi << 1);
      v2f a, bb;
      a.x = sOut[m16 * LSTR + kk];
      a.y = sOut[m16 * LSTR + kk + 1];
      bb.x = WrT[(size_t)kk * TP1 + n];
      bb.y = WrT[(size_t)(kk + 1) * TP1 + n];
      c = wmma4(a, bb, c);
    }
    if (m16 < TP1) {
      float bias = br[t * TP1 + m16];
      #pragma unroll
      for (int r = 0; r < 8; ++r) {
        int m = r + 8 * hi;
        int g = sRow[m];
        float u = (g >= 0) ? route_u[((size_t)hop * BATCH + g) * TP1 + m16]
                           : 0.5f;
        float gum = -logf(-logf(u + 1e-9f) + 1e-9f);
        sLog[m * TP1 + m16] = c[r] + bias + gum;
      }
    }
  }
  __syncthreads();

  // argmax -> route update (first-max tie-break like jnp.argmax)
  if (tid < 16) {
    int g = sRow[tid];
    if (g >= 0) {
      float best = sLog[tid * TP1 + 0];
      int bj = 0;
      for (int j = 1; j < TP1; ++j) {
        float v = sLog[tid * TP1 + j];
        if (v > best) { best = v; bj = j; }
      }
      tempers[g] = bj < (NT_T - 1) ? bj : (NT_T - 1);  // min(sampled, T-1)
      if (bj == NT_T) done[g] = 1;
    }
  }
  // states writeback (in place; each active row owned by exactly one tile)
  for (int idx = tid; idx < 16 * (HID / 4); idx += 128) {
    int m = idx >> 6, c4 = idx & 63;
    int g = sRow[m];
    if (g >= 0)
      *(float4*)(states + (size_t)g * HID + c4 * 4) =
          *(const float4*)&sOut[m * LSTR + c4 * 4];
  }
}

// ---------------------------------------------------------------------------
// 4) out = states @ W_out + b_out   [2048x256]*[256x512]
__global__ __launch_bounds__(128) void proj_out_kernel(
    const float* __restrict__ states, const float* __restrict__ W_out,
    const float* __restrict__ b_out, float* __restrict__ out) {
  __shared__ float sS[16 * LSTR];
  const int tile = blockIdx.x;
  const int tid  = threadIdx.x;
  const int lane = tid & 31, wave = tid >> 5;
  const int m16  = lane & 15, hi = lane >> 4;
#if USE_ASYNC_STAGING
  for (int idx = tid; idx < 16 * (HID / 4); idx += 128) {
    int m = idx >> 6, c4 = idx & 63;
    async_g2l_b128((unsigned)(uintptr_t)&sS[m * LSTR + c4 * 4],
                   states + (size_t)(tile * 16 + m) * HID + c4 * 4);
  }
  wait_async0();
#else
  for (int idx = tid; idx < 16 * (HID / 4); idx += 128) {
    int m = idx >> 6, c4 = idx & 63;
    *(float4*)&sS[m * LSTR + c4 * 4] =
        *(const float4*)(states + (size_t)(tile * 16 + m) * HID + c4 * 4);
  }
#endif
  __syncthreads();
  for (int i = 0; i < 8; ++i) {   // 32 column sub-tiles / 4 waves
    int nt = wave * 8 + i, n = nt * 16 + m16;
    v8f c = {};
    c = gemm16(sS, LSTR, W_out, ODIM, HID, nt, lane, c);
    float bias = b_out[n];
    #pragma unroll
    for (int r = 0; r < 8; ++r)
      out[(size_t)(tile * 16 + r + 8 * hi) * ODIM + n] = c[r] + bias;
  }
}

// ---------------------------------------------------------------------------
extern "C" void kernel_launch(void* const* d_in, const int* in_sizes, int n_in,
                              void* d_out, int out_size, void* d_ws,
                              size_t ws_size, hipStream_t stream) {
  (void)in_sizes; (void)n_in; (void)out_size; (void)ws_size;
  const float* x        = (const float*)d_in[0];
  const int*   init_t   = (const int*)  d_in[1];
  const float* op_noise = (const float*)d_in[2];
  const float* route_u  = (const float*)d_in[3];
  const float* W_in     = (const float*)d_in[4];
  const float* b_in     = (const float*)d_in[5];
  const float* W1       = (const float*)d_in[6];
  const float* b1       = (const float*)d_in[7];
  const float* W2       = (const float*)d_in[8];
  const float* b2       = (const float*)d_in[9];
  const float* Wr       = (const float*)d_in[10];
  const float* br       = (const float*)d_in[11];
  const float* W_out    = (const float*)d_in[12];
  const float* b_out    = (const float*)d_in[13];
  float* out = (float*)d_out;

  // workspace layout (~2.13 MB)
  char* w = (char*)d_ws;
  size_t off = 0;
  float* states   = (float*)(w + off); off += (size_t)BATCH * HID * 4;
  int* tempers    = (int*)(w + off);   off += BATCH * 4;
  int* done       = (int*)(w + off);   off += BATCH * 4;
  int* rowIdx     = (int*)(w + off);   off += BATCH * 4;
  int* counts     = (int*)(w + off);   off += 256;
  int* cursor     = (int*)(w + off);   off += 256;
  int* tileT      = (int*)(w + off);   off += 1024;
  int* tileStart  = (int*)(w + off);   off += 1024;
  int* tileLimit  = (int*)(w + off);   off += 1024;
  int* numTiles   = (int*)(w + off);   off += 256;

  init_state_kernel<<<BATCH / 256, 256, 0, stream>>>(init_t, tempers, done);
  proj_in_kernel<<<BATCH / 16, 128, 0, stream>>>(x, W_in, b_in, states);

  for (int hop = 0; hop < NHOPS; ++hop) {
    bin_zero_kernel<<<1, 32, 0, stream>>>(counts, cursor, numTiles);
    bin_count_kernel<<<BATCH / 256, 256, 0, stream>>>(tempers, done, counts);
    bin_tiles_kernel<<<1, 1, 0, stream>>>(counts, cursor, tileT, tileStart,
                                          tileLimit, numTiles);
    bin_scatter_kernel<<<BATCH / 256, 256, 0, stream>>>(tempers, done, cursor,
                                                        rowIdx);
    expert_kernel<<<MAXTILES, 128, 0, stream>>>(
        W1, b1, W2, b2, Wr, br, op_noise, route_u, states, tempers, done,
        rowIdx, tileT, tileStart, tileLimit, numTiles, hop);
  }

  proj_out_kernel<<<BATCH / 16, 128, 0, stream>>>(states, W_out, b_out, out);
}